// Truncated_28613072126489
// MI455X (gfx1250) — compile-verified
//
#include <hip/hip_runtime.h>

typedef float v2f __attribute__((ext_vector_type(2)));
typedef float v8f __attribute__((ext_vector_type(8)));

#define L_NODES   12288
#define DM        256
#define PROJ      257
#define DK2       514           // 2*PROJ
#define QK_STRIDE 288           // padded row stride (multiple of 32)
#define DEG       32
#define E_TOTAL   (L_NODES * DEG)
#define NSPLIT    16
#define COLS_PER_SPLIT (L_NODES / NSPLIT)   // 768
#define KSTEPS_MAIN    65                    // covers K = 0..259 >= PROJ
#define F_LOG2E   1.44269504088896340736f
#define F_LN2     0.69314718055994530942f
#define NEG_BIG  -3.0e38f

// ---------------------------------------------------------------- zero init
__global__ __launch_bounds__(256) void zero_init_kernel(float* __restrict__ q,
                                                        float* __restrict__ kmat,
                                                        float* __restrict__ rowsum) {
  int i = blockIdx.x * blockDim.x + threadIdx.x;
  if (i < L_NODES * QK_STRIDE) { q[i] = 0.0f; kmat[i] = 0.0f; }
  if (i < L_NODES) rowsum[i] = 0.0f;
}

// ---------------------------------------------------------------- K1: qk = x @ W^T + b  (fp32 WMMA)
// q is stored PRE-SCALED by 16 (== 1/SCALE, exact power of 2), k unscaled.
__global__ __launch_bounds__(32) void qk_gemm_kernel(const float* __restrict__ x,
                                                     const float* __restrict__ W,
                                                     const float* __restrict__ b,
                                                     float* __restrict__ q,
                                                     float* __restrict__ kmat) {
  const int mt   = blockIdx.x;       // 0..767  row tile
  const int nt   = blockIdx.y;       // 0..32   col tile (covers 528 >= 514)
  const int lane = threadIdx.x;
  const int half = lane >> 4;        // 0: K+0..1, 1: K+2..3
  const int l15  = lane & 15;
  const int arow = mt * 16 + l15;    // A-fragment row for this lane
  const int col  = nt * 16 + l15;    // B-fragment col for this lane
  const bool colok = col < DK2;
  const int colc = colok ? col : (DK2 - 1);   // clamped: garbage col never stored

  const float* xrow = x + (size_t)arow * DM;
  const float* wrow = W + (size_t)colc * DM;

  v8f acc = {};
  #pragma unroll 8
  for (int ks = 0; ks < DM / 4; ++ks) {
    const int koff = ks * 4 + half * 2;
    v2f a  = *(const v2f*)(xrow + koff);
    v2f bb = *(const v2f*)(wrow + koff);
    acc = __builtin_amdgcn_wmma_f32_16x16x4_f32(false, a, false, bb,
                                                (short)0, acc, false, false);
  }

  if (colok) {
    const float bias = b[col];
    #pragma unroll
    for (int v = 0; v < 8; ++v) {
      const int r = mt * 16 + v + half * 8;   // C layout: VGPR v -> row v (+8 hi half)
      const float val = acc[v] + bias;
      if (col < PROJ) q[(size_t)r * QK_STRIDE + col]             = val * 16.0f;
      else            kmat[(size_t)r * QK_STRIDE + (col - PROJ)] = val;
    }
  }
}

// ---------------------------------------------------------------- K2: streaming logsumexp of (16q) @ k^T
// Per-lane running (m,s) over the columns this lane owns; one cross-lane
// logsumexp merge at the end of the wave (no shuffles in the block loop).
__global__ __launch_bounds__(32) void scores_lse_kernel(const float* __restrict__ q,
                                                        const float* __restrict__ kmat,
                                                        float* __restrict__ pm,
                                                        float* __restrict__ ps) {
  const int rb   = blockIdx.x;              // 0..191  (64 rows each)
  const int sp   = blockIdx.y;              // 0..NSPLIT-1 column split
  const int lane = threadIdx.x;
  const int half = lane >> 4;
  const int l15  = lane & 15;
  const int row0 = rb * 64;
  const int cbase = sp * COLS_PER_SPLIT;
  const int koffL = half * 2;

  v8f m[4], s[4];
  #pragma unroll
  for (int rt = 0; rt < 4; ++rt)
    #pragma unroll
    for (int v = 0; v < 8; ++v) { m[rt][v] = NEG_BIG; s[rt][v] = 0.0f; }

  #pragma unroll 1   // keep ONE copy of the K-loop + update (avoid 12x code bloat)
  for (int cb = 0; cb < COLS_PER_SPLIT / 64; ++cb) {   // 12 blocks of 64 cols
    const int c0 = cbase + cb * 64;

    v8f acc[4][4];
    #pragma unroll
    for (int rt = 0; rt < 4; ++rt)
      #pragma unroll
      for (int ct = 0; ct < 4; ++ct) acc[rt][ct] = (v8f){};

    for (int ks = 0; ks < KSTEPS_MAIN; ++ks) {
      const int koff = ks * 4 + koffL;
      v2f a[4], bf[4];
      #pragma unroll
      for (int rt = 0; rt < 4; ++rt)
        a[rt] = *(const v2f*)(q + (size_t)(row0 + rt * 16 + l15) * QK_STRIDE + koff);
      #pragma unroll
      for (int ct = 0; ct < 4; ++ct)
        bf[ct] = *(const v2f*)(kmat + (size_t)(c0 + ct * 16 + l15) * QK_STRIDE + koff);
      #pragma unroll
      for (int rt = 0; rt < 4; ++rt)
        #pragma unroll
        for (int ct = 0; ct < 4; ++ct)
          acc[rt][ct] = __builtin_amdgcn_wmma_f32_16x16x4_f32(false, a[rt], false, bf[ct],
                                                              (short)0, acc[rt][ct],
                                                              false, false);
    }

    // per-lane online softmax update (scores already scaled; no shuffles here)
    #pragma unroll
    for (int rt = 0; rt < 4; ++rt) {
      #pragma unroll
      for (int v = 0; v < 8; ++v) {
        const float x0 = acc[rt][0][v];
        const float x1 = acc[rt][1][v];
        const float x2 = acc[rt][2][v];
        const float x3 = acc[rt][3][v];
        const float tm   = fmaxf(fmaxf(x0, x1), fmaxf(x2, x3));
        const float mold = m[rt][v];
        const float mnew = fmaxf(mold, tm);
        const float p = exp2f((x0 - mnew) * F_LOG2E) + exp2f((x1 - mnew) * F_LOG2E)
                      + exp2f((x2 - mnew) * F_LOG2E) + exp2f((x3 - mnew) * F_LOG2E);
        s[rt][v] = s[rt][v] * exp2f((mold - mnew) * F_LOG2E) + p;
        m[rt][v] = mnew;
      }
    }
  }

  // single cross-lane (m,s) logsumexp merge across the 16 lanes of each half
  #pragma unroll
  for (int rt = 0; rt < 4; ++rt) {
    #pragma unroll
    for (int v = 0; v < 8; ++v) {
      float mm = m[rt][v], ss = s[rt][v];
      #pragma unroll
      for (int off = 1; off < 16; off <<= 1) {
        const float om = __shfl_xor(mm, off, 32);   // stays within 16-lane half
        const float os = __shfl_xor(ss, off, 32);
        const float M  = fmaxf(mm, om);
        ss = ss * exp2f((mm - M) * F_LOG2E) + os * exp2f((om - M) * F_LOG2E);
        mm = M;
      }
      m[rt][v] = mm; s[rt][v] = ss;
    }
  }

  if (l15 == 0) {   // lanes 0 and 16 write their half's rows
    #pragma unroll
    for (int rt = 0; rt < 4; ++rt)
      #pragma unroll
      for (int v = 0; v < 8; ++v) {
        const int r = row0 + rt * 16 + v + half * 8;
        pm[sp * L_NODES + r] = m[rt][v];
        ps[sp * L_NODES + r] = s[rt][v];
      }
  }
}

// ---------------------------------------------------------------- K3: merge split partials -> lse
__global__ __launch_bounds__(256) void lse_merge_kernel(const float* __restrict__ pm,
                                                        const float* __restrict__ ps,
                                                        float* __restrict__ lse) {
  int r = blockIdx.x * blockDim.x + threadIdx.x;
  if (r >= L_NODES) return;
  float M = NEG_BIG;
  #pragma unroll
  for (int sp = 0; sp < NSPLIT; ++sp) M = fmaxf(M, pm[sp * L_NODES + r]);
  float S = 0.0f;
  #pragma unroll
  for (int sp = 0; sp < NSPLIT; ++sp)
    S += ps[sp * L_NODES + r] * exp2f((pm[sp * L_NODES + r] - M) * F_LOG2E);
  lse[r] = M + log2f(S) * F_LN2;
}

// ---------------------------------------------------------------- K4: edge softmax values + per-row sum
__global__ __launch_bounds__(256) void edge_vals_kernel(const float* __restrict__ q,
                                                        const float* __restrict__ kmat,
                                                        const int* __restrict__ ei,
                                                        const float* __restrict__ lse,
                                                        float* __restrict__ vals,
                                                        float* __restrict__ rowsum) {
  const int wid  = (int)((blockIdx.x * blockDim.x + threadIdx.x) >> 5);  // one wave per row group
  const int lane = threadIdx.x & 31;
  if (wid >= L_NODES) return;
  const int e0 = wid * DEG;
  const int r  = ei[e0];                    // all 32 edges in group share the row
  const int c  = ei[E_TOTAL + e0 + lane];   // this lane's column

  float qreg[QK_STRIDE / 32];
  #pragma unroll
  for (int t = 0; t < QK_STRIDE / 32; ++t)
    qreg[t] = q[(size_t)r * QK_STRIDE + lane + 32 * t];   // q already x16
  const float lr = lse[r];

  float myval = 0.0f, acc = 0.0f;
  for (int i = 0; i < 32; ++i) {
    const int ci = __shfl(c, i, 32);
    const float* kr = kmat + (size_t)ci * QK_STRIDE;
    float d = 0.0f;
    #pragma unroll
    for (int t = 0; t < QK_STRIDE / 32; ++t)
      d += qreg[t] * kr[lane + 32 * t];
    #pragma unroll
    for (int off = 16; off > 0; off >>= 1)
      d += __shfl_xor(d, off, 32);
    const float val = exp2f((d - lr) * F_LOG2E);
    acc += val;
    if (lane == i) myval = val;
  }
  vals[e0 + lane] = myval;
  if (lane == 0) atomicAdd(&rowsum[r], acc);
}

// ---------------------------------------------------------------- K5: out[e] = vals[e] / rowsum[col[e]]
__global__ __launch_bounds__(256) void edge_out_kernel(const float* __restrict__ vals,
                                                       const float* __restrict__ rowsum,
                                                       const int* __restrict__ ei,
                                                       float* __restrict__ out) {
  int e = blockIdx.x * blockDim.x + threadIdx.x;
  if (e >= E_TOTAL) return;
  out[e] = vals[e] / rowsum[ei[E_TOTAL + e]];   // faithful to column-indexed divisor
}

// ---------------------------------------------------------------- launcher
extern "C" void kernel_launch(void* const* d_in, const int* in_sizes, int n_in,
                              void* d_out, int out_size, void* d_ws, size_t ws_size,
                              hipStream_t stream) {
  (void)in_sizes; (void)n_in; (void)out_size; (void)ws_size;
  const float* x  = (const float*)d_in[0];
  const int*   ei = (const int*)d_in[1];
  const float* W  = (const float*)d_in[2];
  const float* b  = (const float*)d_in[3];
  float* out = (float*)d_out;

  float* ws     = (float*)d_ws;
  float* q      = ws;
  float* kmat   = q    + (size_t)L_NODES * QK_STRIDE;
  float* pm     = kmat + (size_t)L_NODES * QK_STRIDE;
  float* ps     = pm   + (size_t)NSPLIT * L_NODES;
  float* lse    = ps   + (size_t)NSPLIT * L_NODES;
  float* vals   = lse  + (size_t)L_NODES;
  float* rowsum = vals + (size_t)E_TOTAL;

  {
    int n = L_NODES * QK_STRIDE;
    zero_init_kernel<<<(n + 255) / 256, 256, 0, stream>>>(q, kmat, rowsum);
  }
  {
    dim3 grid(L_NODES / 16, (DK2 + 15) / 16);   // 768 x 33
    qk_gemm_kernel<<<grid, 32, 0, stream>>>(x, W, b, q, kmat);
  }
  {
    dim3 grid(L_NODES / 64, NSPLIT);            // 192 x 16
    scores_lse_kernel<<<grid, 32, 0, stream>>>(q, kmat, pm, ps);
  }
  lse_merge_kernel<<<(L_NODES + 255) / 256, 256, 0, stream>>>(pm, ps, lse);
  edge_vals_kernel<<<(L_NODES * 32 + 255) / 256, 256, 0, stream>>>(q, kmat, ei, lse, vals, rowsum);
  edge_out_kernel<<<(E_TOTAL + 255) / 256, 256, 0, stream>>>(vals, rowsum, ei, out);
}